// _CompressJPEG_42992622633224
// MI455X (gfx1250) — compile-verified
//
#include <hip/hip_runtime.h>

typedef __attribute__((ext_vector_type(2))) float v2f;
typedef __attribute__((ext_vector_type(8))) float v8f;

// Correctly-rounded reciprocals of the JPEG quant tables (compile-time consts)
__constant__ float RYTAB[64] = {
    1.f/16, 1.f/11, 1.f/10, 1.f/16, 1.f/24, 1.f/40, 1.f/51, 1.f/61,
    1.f/12, 1.f/12, 1.f/14, 1.f/19, 1.f/26, 1.f/58, 1.f/60, 1.f/55,
    1.f/14, 1.f/13, 1.f/16, 1.f/24, 1.f/40, 1.f/57, 1.f/69, 1.f/56,
    1.f/14, 1.f/17, 1.f/22, 1.f/29, 1.f/51, 1.f/87, 1.f/80, 1.f/62,
    1.f/18, 1.f/22, 1.f/37, 1.f/56, 1.f/68, 1.f/109, 1.f/103, 1.f/77,
    1.f/24, 1.f/35, 1.f/55, 1.f/64, 1.f/81, 1.f/104, 1.f/113, 1.f/92,
    1.f/49, 1.f/64, 1.f/78, 1.f/87, 1.f/103, 1.f/121, 1.f/120, 1.f/101,
    1.f/72, 1.f/92, 1.f/95, 1.f/98, 1.f/112, 1.f/100, 1.f/103, 1.f/99};

__constant__ float RCTAB[64] = {
    1.f/17, 1.f/18, 1.f/24, 1.f/47, 1.f/99, 1.f/99, 1.f/99, 1.f/99,
    1.f/18, 1.f/21, 1.f/26, 1.f/66, 1.f/99, 1.f/99, 1.f/99, 1.f/99,
    1.f/24, 1.f/26, 1.f/56, 1.f/99, 1.f/99, 1.f/99, 1.f/99, 1.f/99,
    1.f/47, 1.f/66, 1.f/99, 1.f/99, 1.f/99, 1.f/99, 1.f/99, 1.f/99,
    1.f/99, 1.f/99, 1.f/99, 1.f/99, 1.f/99, 1.f/99, 1.f/99, 1.f/99,
    1.f/99, 1.f/99, 1.f/99, 1.f/99, 1.f/99, 1.f/99, 1.f/99, 1.f/99,
    1.f/99, 1.f/99, 1.f/99, 1.f/99, 1.f/99, 1.f/99, 1.f/99, 1.f/99,
    1.f/99, 1.f/99, 1.f/99, 1.f/99, 1.f/99, 1.f/99, 1.f/99, 1.f/99};

// DCT-II basis: TBASIS[u*8+x] = cos((2x+1)*u*pi/16)
__constant__ float TBASIS[64] = {
    1.f,          1.f,          1.f,          1.f,          1.f,          1.f,          1.f,          1.f,
    0.98078528f,  0.83146961f,  0.55557023f,  0.19509032f, -0.19509032f, -0.55557023f, -0.83146961f, -0.98078528f,
    0.92387953f,  0.38268343f, -0.38268343f, -0.92387953f, -0.92387953f, -0.38268343f,  0.38268343f,  0.92387953f,
    0.83146961f, -0.19509032f, -0.98078528f, -0.55557023f,  0.55557023f,  0.98078528f,  0.19509032f, -0.83146961f,
    0.70710678f, -0.70710678f, -0.70710678f,  0.70710678f,  0.70710678f, -0.70710678f, -0.70710678f,  0.70710678f,
    0.55557023f, -0.98078528f,  0.19509032f,  0.83146961f, -0.83146961f, -0.19509032f,  0.98078528f, -0.55557023f,
    0.38268343f, -0.92387953f,  0.92387953f, -0.38268343f, -0.38268343f,  0.92387953f, -0.92387953f,  0.38268343f,
    0.19509032f, -0.55557023f,  0.83146961f, -0.98078528f,  0.98078528f, -0.83146961f,  0.55557023f, -0.19509032f};

// Color matrix with *255 folded in (Y) and *255*0.25 pool folded in (chroma)
#define KYR 76.245f
#define KYG 149.685f
#define KYB 29.07f
#define KCBR (-10.756920f)
#define KCBG (-21.118080f)
#define KCBB 31.875f
#define KCRR 31.875f
#define KCRG (-26.691360f)
#define KCRB (-5.183640f)

// Two-stage 16x16 DCT (4 packed 8x8 blocks) via V_WMMA_F32_16X16X4_F32.
// Stage 1: P = Tblk * X      (B operand = X tile from LDS)
// Stage 2: Q = Tblk * P^T    (B operand = P^T via LDS scratch); Q(v,u) = OUT(u,v)
// LDS ops of a wave are in-order (DScnt in-order), so cross-lane RAW through
// the scratch is safe; compiler-only barriers stop IR reordering, and the
// compiler inserts precise s_wait_dscnt for its own load->WMMA deps.
__device__ __forceinline__ v8f dct_tile(const float* xt, int stride,
                                        const v2f a[4], float* ps, int lane) {
  const int n  = lane & 15;
  const int kh = lane >> 4;          // high half-wave handles K+2
  v8f acc = {0.f,0.f,0.f,0.f,0.f,0.f,0.f,0.f};
#pragma unroll
  for (int c = 0; c < 4; ++c) {
    const int k = 4*c + 2*kh;
    v2f bb;
    bb[0] = xt[k * stride + n];
    bb[1] = xt[(k + 1) * stride + n];
    acc = __builtin_amdgcn_wmma_f32_16x16x4_f32(false, a[c], false, bb,
                                                (short)0, acc, false, false);
  }
  // Spill P (D-layout: vgpr j -> row j + 8*kh, col n) to LDS scratch
#pragma unroll
  for (int j = 0; j < 8; ++j)
    ps[(j + 8*kh) * 16 + n] = acc[j];
  asm volatile("" ::: "memory");      // compiler barrier: stores before loads
  v8f q = {0.f,0.f,0.f,0.f,0.f,0.f,0.f,0.f};
#pragma unroll
  for (int c = 0; c < 4; ++c) {
    const int k = 4*c + 2*kh;
    v2f bb;                           // B = P^T : B(k,n) = P(n,k)
    bb[0] = ps[n * 16 + k];
    bb[1] = ps[n * 16 + k + 1];
    q = __builtin_amdgcn_wmma_f32_16x16x4_f32(false, a[c], false, bb,
                                              (short)0, q, false, false);
  }
  asm volatile("" ::: "memory");      // keep ps reuse across tiles ordered
  return q;
}

// q[j] = OUT(u, v=j+8*bC); lane's 8 results are 8 consecutive floats of one
// block. cf[] = DCT_SCALE(u,v)/(table(u,v)*factor), precomputed per lane.
__device__ __forceinline__ void quant_store(v8f q, const float* cf, float* op) {
  float res[8];
#pragma unroll
  for (int j = 0; j < 8; ++j) {
    const float x = q[j] * cf[j];
    const float r = rintf(x);                 // round-half-even == jnp.round
    const float d = x - r;
    res[j] = fmaf(d * d, d, r);               // differentiable round
  }
  *(float4*)(op)     = make_float4(res[0], res[1], res[2], res[3]);
  *(float4*)(op + 4) = make_float4(res[4], res[5], res[6], res[7]);
}

__global__ __launch_bounds__(256)
void jpeg_compress_kernel(const float* __restrict__ xin,
                          const float* __restrict__ factor,
                          float* __restrict__ out) {
  __shared__ float sY[8][1024];   // 32x32 Y' tile per wave
  __shared__ float sC[8][512];    // 16x16 Cb' | 16x16 Cr'
  __shared__ float sP[8][256];    // 16x16 DCT stage-1 scratch

  const int lane = threadIdx.x & 31;
  const int wv   = threadIdx.x >> 5;
  const int wave = blockIdx.x * 8 + wv;     // 8192 waves total
  const int b    = wave >> 8;               // batch
  const int rem  = wave & 255;
  const int ry   = rem >> 4, rx = rem & 15; // 32x32 region in 512x512
  const int r0   = ry * 32, c0 = rx * 32;

  float* ldsY  = sY[wv];
  float* ldsCb = sC[wv];
  float* ldsCr = sC[wv] + 256;
  float* ps    = sP[wv];

  // ---- A operand: Tblk = diag(T,T) in ISA A-layout, from constant table.
  // Lane holds A(m, k) for k in {2kh,2kh+1, 4+2kh,4+2kh+1, 8+..., 12+...};
  // (k&7) repeats between low/high K-halves, so 4 loads + masking suffice.
  const int m  = lane & 15;
  const int kh = lane >> 4;
  const float* tbr = TBASIS + (m & 7) * 8 + 2 * kh;
  const float t0 = tbr[0], t1 = tbr[1], t2 = tbr[4], t3 = tbr[5];
  const bool lo = (m < 8);
  v2f a[4];
  a[0][0] = lo ? t0 : 0.f;  a[0][1] = lo ? t1 : 0.f;
  a[1][0] = lo ? t2 : 0.f;  a[1][1] = lo ? t3 : 0.f;
  a[2][0] = lo ? 0.f : t0;  a[2][1] = lo ? 0.f : t1;
  a[3][0] = lo ? 0.f : t2;  a[3][1] = lo ? 0.f : t3;

  // ---- Per-lane quant coefficient rows (u fixed per lane); reciprocal
  // tables make this pure multiplies — single divide is 1/factor.
  const int u  = lane & 15;
  const int ur = u & 7;
  const float au = (ur == 0) ? 0.70710678118654752f : 1.0f;
  const float invfac = 1.0f / factor[b];
  float cy[8], cc[8];
#pragma unroll
  for (int j = 0; j < 8; ++j) {
    const float aj = (j == 0) ? 0.70710678118654752f : 1.0f;
    const float s = (0.25f * au * aj) * invfac;
    cy[j] = s * RYTAB[ur * 8 + j];
    cc[j] = s * RCTAB[ur * 8 + j];
  }

  // ---- Phase 1: RGB -> (Y-128, Cb', Cr'), pool chroma 2x2, stage in LDS.
  // Each lane handles a 2x4 pixel strip (two quads): 4 iterations, b128 loads.
  const size_t chs = 512u * 512u;
  const float* Rp = xin + (size_t)b * 3u * chs;
#pragma unroll
  for (int it = 0; it < 4; ++it) {
    const int t  = it * 32 + lane;          // 128 strips: 16 rows x 8 cols
    const int qr = t >> 3;                  // quad row 0..15
    const int qp = t & 7;                   // strip col 0..7
    const size_t off = (size_t)(r0 + 2*qr) * 512u + (size_t)(c0 + 4*qp);
    const float4 rA = *(const float4*)(Rp + off);
    const float4 rB = *(const float4*)(Rp + off + 512);
    const float4 gA = *(const float4*)(Rp + chs + off);
    const float4 gB = *(const float4*)(Rp + chs + off + 512);
    const float4 bA = *(const float4*)(Rp + 2*chs + off);
    const float4 bB = *(const float4*)(Rp + 2*chs + off + 512);
    const float r8[8] = {rA.x, rA.y, rA.z, rA.w, rB.x, rB.y, rB.z, rB.w};
    const float g8[8] = {gA.x, gA.y, gA.z, gA.w, gB.x, gB.y, gB.z, gB.w};
    const float b8[8] = {bA.x, bA.y, bA.z, bA.w, bB.x, bB.y, bB.z, bB.w};
    float y8[8];
#pragma unroll
    for (int p = 0; p < 8; ++p)             // Y' = Y - 128 folded in
      y8[p] = fmaf(KYB, b8[p], fmaf(KYG, g8[p], fmaf(KYR, r8[p], -128.f)));
    // Quad RGB sums (quad0 = cols 0-1, quad1 = cols 2-3), shared by Cb & Cr
    const float sr0 = (r8[0]+r8[1]) + (r8[4]+r8[5]);
    const float sg0 = (g8[0]+g8[1]) + (g8[4]+g8[5]);
    const float sb0 = (b8[0]+b8[1]) + (b8[4]+b8[5]);
    const float sr1 = (r8[2]+r8[3]) + (r8[6]+r8[7]);
    const float sg1 = (g8[2]+g8[3]) + (g8[6]+g8[7]);
    const float sb1 = (b8[2]+b8[3]) + (b8[6]+b8[7]);
    const float cb0 = fmaf(KCBB, sb0, fmaf(KCBG, sg0, KCBR * sr0));
    const float cb1 = fmaf(KCBB, sb1, fmaf(KCBG, sg1, KCBR * sr1));
    const float cr0 = fmaf(KCRB, sb0, fmaf(KCRG, sg0, KCRR * sr0));
    const float cr1 = fmaf(KCRB, sb1, fmaf(KCRG, sg1, KCRR * sr1));
    const int ly = 2*qr, lx = 4*qp;
    *(float4*)(ldsY + ly*32 + lx)       = make_float4(y8[0], y8[1], y8[2], y8[3]);
    *(float4*)(ldsY + (ly+1)*32 + lx)   = make_float4(y8[4], y8[5], y8[6], y8[7]);
    *(float2*)(ldsCb + qr*16 + 2*qp)    = make_float2(cb0, cb1);
    *(float2*)(ldsCr + qr*16 + 2*qp)    = make_float2(cr0, cr1);
  }
  asm volatile("" ::: "memory");   // LDS in-order per wave; stop IR reordering

  // ---- Phase 2: WMMA DCT + quantize + store
  const int bC = lane >> 4;        // output block-column within 16x16 tile

  float* outY  = out;
  float* outCb = out + (size_t)32 * 4096 * 64;
  float* outCr = outCb + (size_t)32 * 1024 * 64;

#pragma unroll
  for (int s = 0; s < 4; ++s) {    // four 16x16 Y subtiles
    const int ty0 = (s >> 1) * 16, tx0 = (s & 1) * 16;
    v8f qd = dct_tile(ldsY + ty0*32 + tx0, 32, a, ps, lane);
    const int br = ((r0 + ty0) >> 3) + (u >> 3);
    const int bc = ((c0 + tx0) >> 3) + bC;
    quant_store(qd, cy, outY + ((size_t)(b*4096 + br*64 + bc)) * 64 + ur*8);
  }
  {
    v8f qd = dct_tile(ldsCb, 16, a, ps, lane);
    const int br = ry*2 + (u >> 3), bc = rx*2 + bC;
    quant_store(qd, cc, outCb + ((size_t)(b*1024 + br*32 + bc)) * 64 + ur*8);
  }
  {
    v8f qd = dct_tile(ldsCr, 16, a, ps, lane);
    const int br = ry*2 + (u >> 3), bc = rx*2 + bC;
    quant_store(qd, cc, outCr + ((size_t)(b*1024 + br*32 + bc)) * 64 + ur*8);
  }
}

extern "C" void kernel_launch(void* const* d_in, const int* in_sizes, int n_in,
                              void* d_out, int out_size, void* d_ws, size_t ws_size,
                              hipStream_t stream) {
  const float* x      = (const float*)d_in[0];
  const float* factor = (const float*)d_in[1];
  float* out = (float*)d_out;
  // 8192 waves (32 batches x 16x16 regions), 8 waves per 256-thread block
  jpeg_compress_kernel<<<dim3(1024), dim3(256), 0, stream>>>(x, factor, out);
}